// Net_41360535060938
// MI455X (gfx1250) — compile-verified
//
#include <hip/hip_runtime.h>
#include <math.h>

#define HH   256
#define OUTD 40
#define CAT  (2*HH + OUTD)   // 552

typedef __attribute__((ext_vector_type(2))) float v2f;
typedef __attribute__((ext_vector_type(8))) float v8f;

// ---------------- degree / normalization ----------------

__global__ void set_ones_kernel(float* __restrict__ p, int n) {
    int i = blockIdx.x * blockDim.x + threadIdx.x;
    if (i < n) p[i] = 1.0f;                    // self-loop contributes 1 to deg
}

__global__ void deg_count_kernel(const int* __restrict__ col, float* __restrict__ deg, int E) {
    int e = blockIdx.x * blockDim.x + threadIdx.x;
    if (e < E) atomicAdd(&deg[col[e]], 1.0f);
}

__global__ void rsqrt_kernel(float* __restrict__ p, int n) {
    int i = blockIdx.x * blockDim.x + threadIdx.x;
    if (i < n) p[i] = rsqrtf(p[i]);
}

// ---------------- fp32 WMMA GEMM: T = X(MxK) @ W(KxNOUT) ----------------
// One wave computes a 16 x (16*NT) output strip with V_WMMA_F32_16X16X4_F32.
// K and NOUT are compile-time so all inner-loop addresses fold into the
// global_load 24-bit immediate offset (no per-step VALU address math).
// Out-of-range rows/cols handled by clamping base addresses (clamped garbage
// only reaches accumulator slots that are never stored).
// A 16x4 f32 layout: lanes 0-15 -> K={k0,k0+1}, lanes 16-31 -> K={k0+2,k0+3}.
// B mirrored on N. C/D: VGPR r -> row half*8+r, col lane%16.

template <int NT, int K, int NOUT>
__global__ __launch_bounds__(32) void gemm_f32_wmma(
    const float* __restrict__ X, const float* __restrict__ W,
    float* __restrict__ T, int M, int ngroups)
{
    const int tile = blockIdx.x;
    const int mt   = tile / ngroups;
    const int ng   = tile % ngroups;
    const int lane = threadIdx.x & 31;
    const int half = lane >> 4;     // 0 or 1
    const int l16  = lane & 15;

    int m = mt * 16 + l16;
    if (m > M - 1) m = M - 1;                       // clamp: rows >= M never stored
    // per-lane A base: row m, K-offset half*2 (b64 loads, 8B aligned)
    const float* __restrict__ arow = X + (size_t)m * K + half * 2;

    // per-lane B base pointers: column ncl[j], K-offset half*2
    const float* bcol[NT];
#pragma unroll
    for (int j = 0; j < NT; ++j) {
        int n = ng * (16 * NT) + 16 * j + l16;
        if (n > NOUT - 1) n = NOUT - 1;             // clamp: cols >= NOUT never stored
        bcol[j] = W + (size_t)(half * 2) * NOUT + n;
    }

    v8f acc[NT];
#pragma unroll
    for (int j = 0; j < NT; ++j) acc[j] = (v8f){0.f,0.f,0.f,0.f,0.f,0.f,0.f,0.f};

#pragma unroll 8
    for (int k0 = 0; k0 < K; k0 += 4) {
        const v2f a = *(const v2f*)(arow + k0);     // global_load_b64, imm offset k0*4
#pragma unroll
        for (int j = 0; j < NT; ++j) {
            v2f b;
            b.x = bcol[j][(size_t)k0 * NOUT];        // imm offset k0*NOUT*4
            b.y = bcol[j][(size_t)k0 * NOUT + NOUT]; // imm offset (k0+1)*NOUT*4
            acc[j] = __builtin_amdgcn_wmma_f32_16x16x4_f32(
                /*neg_a=*/false, a, /*neg_b=*/false, b,
                /*c_mod=*/(short)0, acc[j], /*reuse_a=*/false, /*reuse_b=*/false);
        }
    }

#pragma unroll
    for (int j = 0; j < NT; ++j) {
        const int n = ng * (16 * NT) + 16 * j + l16;
        if (n < NOUT) {
#pragma unroll
            for (int r = 0; r < 8; ++r) {
                int mr = mt * 16 + half * 8 + r;
                if (mr < M) T[(size_t)mr * NOUT + n] = acc[j][r];
            }
        }
    }
}

// ---------------- edge scatter: agg[col] += t[row] * dis[row]*dis[col] ----------------
// One wave per edge; b128 gather loads + fp32 global atomics. F4 = F/4.

__global__ void scatter_edges_kernel(
    const int* __restrict__ row, const int* __restrict__ col,
    const float* __restrict__ dis, const float* __restrict__ t,
    float* __restrict__ agg, int E, int F4)
{
    const int wave = blockIdx.x * (blockDim.x >> 5) + (threadIdx.x >> 5);
    const int lane = threadIdx.x & 31;
    if (wave >= E) return;
    const int r = row[wave];
    const int c = col[wave];
    const float nrm = dis[r] * dis[c];
    const float4* __restrict__ src = (const float4*)(t + (size_t)r * F4 * 4);
    float*        __restrict__ dst = agg + (size_t)c * F4 * 4;
    for (int f = lane; f < F4; f += 32) {
        float4 v = src[f];
        atomicAdd(dst + 4*f + 0, v.x * nrm);
        atomicAdd(dst + 4*f + 1, v.y * nrm);
        atomicAdd(dst + 4*f + 2, v.z * nrm);
        atomicAdd(dst + 4*f + 3, v.w * nrm);
    }
}

// ---------------- epilogue: h = relu(agg + t*dis^2 (self loop) + b) ----------------

__global__ void bias_relu_self_kernel(
    float4* __restrict__ h, const float4* __restrict__ t,
    const float* __restrict__ dis, const float4* __restrict__ b,
    int Nn, int F4)
{
    long long i = (long long)blockIdx.x * blockDim.x + threadIdx.x;
    long long total = (long long)Nn * F4;
    if (i >= total) return;
    int node = (int)(i / F4);
    int f    = (int)(i % F4);
    float d = dis[node];
    float d2 = d * d;
    float4 hv = h[i], tv = t[i], bv = b[f];
    hv.x = fmaxf(fmaf(tv.x, d2, hv.x) + bv.x, 0.f);
    hv.y = fmaxf(fmaf(tv.y, d2, hv.y) + bv.y, 0.f);
    hv.z = fmaxf(fmaf(tv.z, d2, hv.z) + bv.z, 0.f);
    hv.w = fmaxf(fmaf(tv.w, d2, hv.w) + bv.w, 0.f);
    h[i] = hv;
}

// ---------------- final: out = log_softmax(concat(h1,h2,h3) @ lin_w + lin_b) ----------------
// One 64-thread block per node; 552 features staged in LDS.

__global__ __launch_bounds__(64) void final_lin_logsoftmax_kernel(
    const float* __restrict__ h1, const float* __restrict__ h2,
    const float* __restrict__ h3,
    const float* __restrict__ lw, const float* __restrict__ lb,
    float* __restrict__ out, int Nn)
{
    __shared__ float c[CAT];
    __shared__ float z[OUTD];
    __shared__ float red[2];

    const int node = blockIdx.x;
    const int t = threadIdx.x;
    if (node >= Nn) return;

    for (int k = t; k < HH;   k += 64) c[k]        = h1[(size_t)node * HH + k];
    for (int k = t; k < HH;   k += 64) c[HH + k]   = h2[(size_t)node * HH + k];
    for (int k = t; k < OUTD; k += 64) c[2*HH + k] = h3[(size_t)node * OUTD + k];
    __syncthreads();

    if (t < OUTD) {
        float acc = lb[t];
        for (int k = 0; k < CAT; ++k)
            acc = fmaf(c[k], lw[(size_t)k * OUTD + t], acc);
        z[t] = acc;
    }
    __syncthreads();

    if (t == 0) {
        float mx = z[0];
        for (int j = 1; j < OUTD; ++j) mx = fmaxf(mx, z[j]);
        float s = 0.f;
        for (int j = 0; j < OUTD; ++j) s += expf(z[j] - mx);
        red[0] = mx;
        red[1] = logf(s);
    }
    __syncthreads();

    if (t < OUTD) out[(size_t)node * OUTD + t] = z[t] - red[0] - red[1];
}

// ---------------- launch ----------------

extern "C" void kernel_launch(void* const* d_in, const int* in_sizes, int n_in,
                              void* d_out, int out_size, void* d_ws, size_t ws_size,
                              hipStream_t stream)
{
    const float* x  = (const float*)d_in[0];
    const int*   ei = (const int*)  d_in[1];
    const float* W1 = (const float*)d_in[2];
    const float* b1 = (const float*)d_in[3];
    const float* W2 = (const float*)d_in[4];
    const float* b2 = (const float*)d_in[5];
    const float* W3 = (const float*)d_in[6];
    const float* b3 = (const float*)d_in[7];
    const float* lw = (const float*)d_in[8];
    const float* lb = (const float*)d_in[9];
    float* out = (float*)d_out;

    const int N = in_sizes[0] / HH;   // nodes
    const int E = in_sizes[1] / 2;    // edges
    const int* row = ei;              // edge_index[0]
    const int* col = ei + E;          // edge_index[1]

    // workspace layout (floats)
    float* dis = (float*)d_ws;                 // N       (deg -> rsqrt(deg))
    float* t   = dis + N;                      // N*HH    (transformed features)
    float* h1  = t   + (size_t)N * HH;         // N*HH
    float* h2  = h1  + (size_t)N * HH;         // N*HH
    float* h3  = h2  + (size_t)N * HH;         // N*OUTD

    const int T256 = 256;
    // degrees + symmetric normalization
    set_ones_kernel <<<(N + T256 - 1) / T256, T256, 0, stream>>>(dis, N);
    deg_count_kernel<<<(E + T256 - 1) / T256, T256, 0, stream>>>(col, dis, E);
    rsqrt_kernel    <<<(N + T256 - 1) / T256, T256, 0, stream>>>(dis, N);

    const int mtiles   = (N + 15) / 16;          // 6250
    const int ngroupsH = HH / 64;                // 4 groups of 64 cols (NT=4)
    const int wavesPB  = T256 / 32;              // 8 edges per block in scatter
    const int edgeBlk  = (E + wavesPB - 1) / wavesPB;

    // ---- layer 1: h1 = relu(prop(x @ W1) + b1) ----
    gemm_f32_wmma<4, HH, HH><<<mtiles * ngroupsH, 32, 0, stream>>>(x, W1, t, N, ngroupsH);
    hipMemsetAsync(h1, 0, (size_t)N * HH * sizeof(float), stream);
    scatter_edges_kernel<<<edgeBlk, T256, 0, stream>>>(row, col, dis, t, h1, E, HH/4);
    bias_relu_self_kernel<<<(int)(((long long)N * (HH/4) + T256 - 1) / T256), T256, 0, stream>>>(
        (float4*)h1, (const float4*)t, dis, (const float4*)b1, N, HH/4);

    // ---- layer 2: h2 = relu(prop(h1 @ W2) + b2) ----
    gemm_f32_wmma<4, HH, HH><<<mtiles * ngroupsH, 32, 0, stream>>>(h1, W2, t, N, ngroupsH);
    hipMemsetAsync(h2, 0, (size_t)N * HH * sizeof(float), stream);
    scatter_edges_kernel<<<edgeBlk, T256, 0, stream>>>(row, col, dis, t, h2, E, HH/4);
    bias_relu_self_kernel<<<(int)(((long long)N * (HH/4) + T256 - 1) / T256), T256, 0, stream>>>(
        (float4*)h2, (const float4*)t, dis, (const float4*)b2, N, HH/4);

    // ---- layer 3: h3 = relu(prop(h2 @ W3) + b3), width 40 (one NT=3 group: 48 padded) ----
    gemm_f32_wmma<3, HH, OUTD><<<mtiles, 32, 0, stream>>>(h2, W3, t, N, 1);
    hipMemsetAsync(h3, 0, (size_t)N * OUTD * sizeof(float), stream);
    scatter_edges_kernel<<<edgeBlk, T256, 0, stream>>>(row, col, dis, t, h3, E, OUTD/4);
    bias_relu_self_kernel<<<(int)(((long long)N * (OUTD/4) + T256 - 1) / T256), T256, 0, stream>>>(
        (float4*)h3, (const float4*)t, dis, (const float4*)b3, N, OUTD/4);

    // ---- final projection + log_softmax ----
    final_lin_logsoftmax_kernel<<<N, 64, 0, stream>>>(h1, h2, h3, lw, lb, out, N);
}